// MyModel_87522843560611
// MI455X (gfx1250) — compile-verified
//
#include <hip/hip_runtime.h>

typedef __attribute__((ext_vector_type(2))) float v2f;
typedef __attribute__((ext_vector_type(8))) float v8f;

#define UNITS 10
#define T_IN  50
#define T_OUT 3

// lane l <-> lane l^16 exchange (SWAPX16): xor=0x10, or=0, and=0x1f -> 0x401f
__device__ __forceinline__ float swap16(float x) {
  return __int_as_float(__builtin_amdgcn_ds_swizzle(__float_as_int(x), 0x401f));
}

__device__ __forceinline__ float sigf(float x) {
  return __builtin_amdgcn_rcpf(1.0f + __expf(-x));
}

// D = A(16x4 f32) x B(4x16 f32) + C ; A lane l: m=l%16, k=p+2*(l>=16); B lane l: n=l%16, k=p+2*(l>=16)
__device__ __forceinline__ v8f wmma4(float a0, float a1, float b0, float b1, v8f c) {
  v2f a = {a0, a1};
  v2f b = {b0, b1};
  return __builtin_amdgcn_wmma_f32_16x16x4_f32(false, a, false, b, (short)0, c, false, false);
}

__global__ __launch_bounds__(256) void lstm_wmma_gfx1250(
    const float* __restrict__ X,
    const float* __restrict__ W1, const float* __restrict__ R1, const float* __restrict__ b1,
    const float* __restrict__ W2, const float* __restrict__ R2, const float* __restrict__ b2,
    const float* __restrict__ Wd, const float* __restrict__ bd,
    float* __restrict__ Out, int Btotal)
{
  const int lane = threadIdx.x & 31;
  const int col  = lane & 15;      // batch-in-tile (N) / weight row m (M)
  const int hi   = lane >> 4;      // half select
  const int wave = (blockIdx.x * blockDim.x + threadIdx.x) >> 5;
  const int b0   = wave * 16;
  if (b0 >= Btotal) return;

  // ---- A operands for LSTM1: rows 0..9 = R1, row 10 = W1, row 11 = b1 (gate tile t, kchunk j) ----
  float A1r[4][3][2];
#pragma unroll
  for (int t = 0; t < 4; ++t)
#pragma unroll
    for (int j = 0; j < 3; ++j)
#pragma unroll
      for (int p = 0; p < 2; ++p) {
        const int k = 4 * j + p + 2 * hi;   // 0..11
        const int m = col;
        float v = 0.0f;
        if (m < UNITS) {
          const int g = t * UNITS + m;
          if (k < UNITS)       v = R1[k * 40 + g];
          else if (k == 10)    v = W1[g];
          else                 v = b1[g];
        }
        A1r[t][j][p] = v;
      }

  // ---- A operands for LSTM2: rows 0..9 = R2, 10..19 = W2 (input = h1), 20 = b2, 21..23 = 0 ----
  float A2r[4][6][2];
#pragma unroll
  for (int t = 0; t < 4; ++t)
#pragma unroll
    for (int j = 0; j < 6; ++j)
#pragma unroll
      for (int p = 0; p < 2; ++p) {
        const int k = 4 * j + p + 2 * hi;   // 0..23
        const int m = col;
        float v = 0.0f;
        if (m < UNITS) {
          const int g = t * UNITS + m;
          if (k < 10)        v = R2[k * 40 + g];
          else if (k < 20)   v = W2[(k - 10) * 40 + g];
          else if (k == 20)  v = b2[g];
        }
        A2r[t][j][p] = v;
      }

  // dense head weights, D-layout (unit = v + 8*hi)
  float wdv[8];
#pragma unroll
  for (int v = 0; v < 8; ++v) {
    const int u = v + 8 * hi;
    wdv[v] = (u < UNITS) ? Wd[u] : 0.0f;
  }
  const float bdv = bd[0];

  const float* Xrow = X + (size_t)(b0 + col) * T_IN;

  // ================= LSTM1 (50 steps) =================
  float h[8], c[8];
#pragma unroll
  for (int v = 0; v < 8; ++v) { h[v] = 0.0f; c[v] = 0.0f; }

  for (int t = 0; t < T_IN; ++t) {
    const float xt = Xrow[t];

    // B chunks: k 0..9 = h units, k=10 -> x_t, k=11 -> 1.0 (bias row)
    float Bc[3][2];
    Bc[0][0] = hi ? swap16(h[2]) : h[0];
    Bc[0][1] = hi ? swap16(h[3]) : h[1];
    Bc[1][0] = hi ? swap16(h[6]) : h[4];
    Bc[1][1] = hi ? swap16(h[7]) : h[5];
    Bc[2][0] = hi ? xt           : swap16(h[0]);   // k=8 | k=10
    Bc[2][1] = hi ? 1.0f         : swap16(h[1]);   // k=9 | k=11

    v8f z[4];
#pragma unroll
    for (int tt = 0; tt < 4; ++tt) {
      v8f acc = {0.f,0.f,0.f,0.f,0.f,0.f,0.f,0.f};
#pragma unroll
      for (int j = 0; j < 3; ++j)
        acc = wmma4(A1r[tt][j][0], A1r[tt][j][1], Bc[j][0], Bc[j][1], acc);
      z[tt] = acc;
    }

#pragma unroll
    for (int v = 0; v < 8; ++v) {
      const float iv = sigf(z[0][v]);
      const float fv = sigf(z[1][v]);
      const float gv = fmaxf(z[2][v], 0.0f);
      const float ov = sigf(z[3][v]);
      c[v] = fv * c[v] + iv * gv;
      h[v] = ov * fmaxf(c[v], 0.0f);
    }
  }

  // ================= LSTM2 (3 steps, input = h1 each step) =================
  const float h1s0 = swap16(h[0]);
  const float h1s1 = swap16(h[1]);
  const float B3_0 = hi ? swap16(h[4]) : h[2];   // k=12 | k=14 (h1 u2|u4)
  const float B3_1 = hi ? swap16(h[5]) : h[3];   // k=13 | k=15 (h1 u3|u5)
  const float B4_0 = hi ? h[0] : h[6];           // k=16 | k=18 (h1 u6|u8)
  const float B4_1 = hi ? h[1] : h[7];           // k=17 | k=19 (h1 u7|u9)
  const float B5_0 = hi ? 0.0f : 1.0f;           // k=20 bias row | k=22 pad

  float h2[8], c2[8];
#pragma unroll
  for (int v = 0; v < 8; ++v) { h2[v] = 0.0f; c2[v] = 0.0f; }

  for (int tau = 0; tau < T_OUT; ++tau) {
    float Bc[6][2];
    Bc[0][0] = hi ? swap16(h2[2]) : h2[0];
    Bc[0][1] = hi ? swap16(h2[3]) : h2[1];
    Bc[1][0] = hi ? swap16(h2[6]) : h2[4];
    Bc[1][1] = hi ? swap16(h2[7]) : h2[5];
    Bc[2][0] = hi ? h1s0 : swap16(h2[0]);  // k=8 (h2 u8) | k=10 (h1 u0)
    Bc[2][1] = hi ? h1s1 : swap16(h2[1]);  // k=9 (h2 u9) | k=11 (h1 u1)
    Bc[3][0] = B3_0;  Bc[3][1] = B3_1;
    Bc[4][0] = B4_0;  Bc[4][1] = B4_1;
    Bc[5][0] = B5_0;  Bc[5][1] = 0.0f;

    v8f z[4];
#pragma unroll
    for (int tt = 0; tt < 4; ++tt) {
      v8f acc = {0.f,0.f,0.f,0.f,0.f,0.f,0.f,0.f};
#pragma unroll
      for (int j = 0; j < 6; ++j)
        acc = wmma4(A2r[tt][j][0], A2r[tt][j][1], Bc[j][0], Bc[j][1], acc);
      z[tt] = acc;
    }

#pragma unroll
    for (int v = 0; v < 8; ++v) {
      const float iv = sigf(z[0][v]);
      const float fv = sigf(z[1][v]);
      const float gv = fmaxf(z[2][v], 0.0f);
      const float ov = sigf(z[3][v]);
      c2[v] = fv * c2[v] + iv * gv;
      h2[v] = ov * fmaxf(c2[v], 0.0f);
    }

    // dense head: out[b,tau] = sum_u h2[u]*Wd[u] + bd
    float part = 0.0f;
#pragma unroll
    for (int v = 0; v < 8; ++v) part += h2[v] * wdv[v];
    const float tot = part + swap16(part);
    if (hi == 0) Out[(size_t)(b0 + col) * T_OUT + tau] = tot + bdv;
  }
}

extern "C" void kernel_launch(void* const* d_in, const int* in_sizes, int n_in,
                              void* d_out, int out_size, void* d_ws, size_t ws_size,
                              hipStream_t stream) {
  const float* X  = (const float*)d_in[0];
  const float* W1 = (const float*)d_in[1];
  const float* R1 = (const float*)d_in[2];
  const float* b1 = (const float*)d_in[3];
  const float* W2 = (const float*)d_in[4];
  const float* R2 = (const float*)d_in[5];
  const float* b2 = (const float*)d_in[6];
  const float* Wd = (const float*)d_in[7];
  const float* bd = (const float*)d_in[8];
  float* Out = (float*)d_out;

  const int B = in_sizes[0] / T_IN;
  const int waves = (B + 15) / 16;
  const int wavesPerBlock = 8;                 // 256 threads = 8 wave32
  const int grid = (waves + wavesPerBlock - 1) / wavesPerBlock;
  lstm_wmma_gfx1250<<<grid, 256, 0, stream>>>(X, W1, R1, b1, W2, R2, b2, Wd, bd, Out, B);
}